// MultiHeadSelfAttentionDecoderCached_35613868818467
// MI455X (gfx1250) — compile-verified
//
#include <hip/hip_runtime.h>
#include <math.h>

// ---------------------------------------------------------------------------
// MHA decoder step with KV cache, fp32.  B=16, S=1, D=2048, H=16, L=4096.
// Memory-bound (~2.2GB => ~95us floor @ 23.3TB/s).
//   * Projections (16x2048x2048) on V_WMMA_F32_16X16X4_F32 (exact fp32).
//   * Attention fuses the cache->output K/V copy with score / p*v passes so
//     the 536MB K/V stream touches HBM exactly once, with NT cache hints.
// ---------------------------------------------------------------------------

typedef __attribute__((ext_vector_type(2))) float v2f;
typedef __attribute__((ext_vector_type(4))) float v4f;
typedef __attribute__((ext_vector_type(8))) float v8f;

#define D_MODEL 2048
#define NHEADS  16
#define LCACHE  4096
#define DKH     128
#define BATCH   16

// ---------------------------------------------------------------------------
// One wave computes one 16x16 tile of C = A(16x2048) @ W(2048x2048) + bias.
// mode 0: C[m][n] -> out[m*2048 + n]
// mode 1: C[m][n] -> out[((m*16 + n/128)*4097 + 4096)*128 + n%128]
// Both are affine in m: dst = base + m*mstride  (branch-free epilogue).
// ---------------------------------------------------------------------------
__device__ __forceinline__ void gemm16_tile(
    const float* __restrict__ A,
    const float* __restrict__ W,
    const float* __restrict__ bias,
    float* __restrict__ out,
    int mode, int tileIdx, int wave, int lane)
{
    const int n0   = (tileIdx * 4 + wave) * 16;
    const int half = lane >> 4;        // 0..1
    const int mn   = lane & 15;        // M for A, N for B/C/D

    // A layout (16x4 f32): lane m = mn; VGPR v holds K = v + 2*half
    const float* aRow = A + (size_t)mn * D_MODEL + 2 * half;
    // B layout (4x16 f32): lane n = mn; VGPR v holds K = v + 2*half
    const float* bCol = W + n0 + mn;

    v8f acc = {};
#pragma unroll 8
    for (int k = 0; k < D_MODEL; k += 4) {
        v2f a, b;
        a.x = aRow[k];
        a.y = aRow[k + 1];
        const float* bp = bCol + (size_t)(k + 2 * half) * D_MODEL;
        b.x = bp[0];
        b.y = bp[D_MODEL];
        acc = __builtin_amdgcn_wmma_f32_16x16x4_f32(
            /*neg_a=*/false, a, /*neg_b=*/false, b,
            /*c_mod=*/(short)0, acc, /*reuse_a=*/false, /*reuse_b=*/false);
    }

    // D layout (16x16 f32): lane holds N = mn, VGPR v holds M = v + 8*half
    const int   n    = n0 + mn;
    const float bval = bias[n];
    size_t base, mstride;
    if (mode == 0) {
        base    = (size_t)n;
        mstride = D_MODEL;
    } else {
        base    = ((size_t)(n >> 7) * (LCACHE + 1) + LCACHE) * DKH + (n & 127);
        mstride = (size_t)NHEADS * (LCACHE + 1) * DKH;
    }
    float* dst = out + base + (size_t)(8 * half) * mstride;
#pragma unroll
    for (int v = 0; v < 8; ++v)
        dst[(size_t)v * mstride] = acc[v] + bval;
}

// Fused Q/K/V projection: grid.x = 96; blockIdx.x>>5 selects matrix.
__global__ __launch_bounds__(128) void qkv_proj_wmma(
    const float* __restrict__ x,
    const float* __restrict__ Wq, const float* __restrict__ bq,
    const float* __restrict__ Wk, const float* __restrict__ bk,
    const float* __restrict__ Wv, const float* __restrict__ bv,
    float* __restrict__ Qws, float* __restrict__ Kout, float* __restrict__ Vout)
{
    const int idx  = blockIdx.x >> 5;          // 0=Q, 1=K, 2=V
    const int tile = blockIdx.x & 31;
    const int wave = threadIdx.x >> 5;
    const int lane = threadIdx.x & 31;

    const float* W    = (idx == 0) ? Wq : (idx == 1) ? Wk : Wv;
    const float* bias = (idx == 0) ? bq : (idx == 1) ? bk : bv;
    float*       out  = (idx == 0) ? Qws : (idx == 1) ? Kout : Vout;
    const int    mode = (idx == 0) ? 0 : 1;

    gemm16_tile(x, W, bias, out, mode, tile, wave, lane);
}

// Output projection: grid.x = 32.
__global__ __launch_bounds__(128) void oproj_wmma(
    const float* __restrict__ A,
    const float* __restrict__ W,
    const float* __restrict__ bias,
    float* __restrict__ out)
{
    gemm16_tile(A, W, bias, out, 0, blockIdx.x, threadIdx.x >> 5, threadIdx.x & 31);
}

// ---------------------------------------------------------------------------
// Fused attention: one block per (b,h).  512 threads = 16 waves.
// Phase 1: stream K rows (NT copy cache->Kout) + q.k dot -> scores in LDS
// Phase 2: block softmax over 4097 scores
// Phase 3: stream V rows (NT copy cache->Vout) + p*v accumulation
// Row 4096 (new token) is read back from Kout/Vout (written by qkv_proj).
// ---------------------------------------------------------------------------
#define ATTN_WAVES 16
__global__ __launch_bounds__(32 * ATTN_WAVES) void attn_fused(
    const float* __restrict__ cache_k,
    const float* __restrict__ cache_v,
    const float* __restrict__ Qws,      // 16 x 2048
    float* __restrict__ Kout,           // 16x16x4097x128
    float* __restrict__ Vout,
    float* __restrict__ attn_ws)        // 16 x 2048
{
    const int bh   = blockIdx.x;        // 0..255
    const int b    = bh >> 4;
    const int h    = bh & 15;
    const int tid  = threadIdx.x;
    const int wave = tid >> 5;
    const int lane = tid & 31;

    __shared__ __align__(16) float s_scores[LCACHE + 1];
    __shared__ __align__(16) float s_q[DKH];
    __shared__ __align__(16) float s_acc[ATTN_WAVES][DKH];
    __shared__ float s_redA[ATTN_WAVES];
    __shared__ float s_redB[ATTN_WAVES];
    __shared__ float s_stat[2];         // [0]=max, [1]=1/sum

    if (tid < DKH) s_q[tid] = Qws[(size_t)b * D_MODEL + h * DKH + tid];
    __syncthreads();

    const v4f qf = reinterpret_cast<const v4f*>(s_q)[lane];
    const size_t srcBase = (size_t)bh * LCACHE * DKH;
    const size_t dstBase = (size_t)bh * (LCACHE + 1) * DKH;
    const float  scale   = 0.08838834764831845f;   // 1/sqrt(128)

    // ---- Phase 1: K copy (non-temporal) + scores, one row per wave ----
    for (int k = wave; k <= LCACHE; k += ATTN_WAVES) {
        v4f kr;
        if (k < LCACHE) {
            kr = __builtin_nontemporal_load(
                     reinterpret_cast<const v4f*>(cache_k + srcBase + (size_t)k * DKH) + lane);
            __builtin_nontemporal_store(
                kr, reinterpret_cast<v4f*>(Kout + dstBase + (size_t)k * DKH) + lane);
        } else {
            kr = reinterpret_cast<const v4f*>(Kout + dstBase + (size_t)k * DKH)[lane];
        }
        float d = kr.x * qf.x + kr.y * qf.y + kr.z * qf.z + kr.w * qf.w;
#pragma unroll
        for (int off = 16; off > 0; off >>= 1)
            d += __shfl_down(d, off, 32);
        if (lane == 0) s_scores[k] = d * scale;
    }
    __syncthreads();

    // ---- Phase 2: softmax ----
    float mx = -INFINITY;
    for (int k = tid; k <= LCACHE; k += 32 * ATTN_WAVES) mx = fmaxf(mx, s_scores[k]);
#pragma unroll
    for (int off = 16; off > 0; off >>= 1)
        mx = fmaxf(mx, __shfl_down(mx, off, 32));
    if (lane == 0) s_redA[wave] = mx;
    __syncthreads();
    if (tid == 0) {
        float m0 = s_redA[0];
#pragma unroll
        for (int i = 1; i < ATTN_WAVES; ++i) m0 = fmaxf(m0, s_redA[i]);
        s_stat[0] = m0;
    }
    __syncthreads();
    mx = s_stat[0];

    float sum = 0.0f;
    for (int k = tid; k <= LCACHE; k += 32 * ATTN_WAVES) {
        const float p = __expf(s_scores[k] - mx);
        s_scores[k] = p;
        sum += p;
    }
#pragma unroll
    for (int off = 16; off > 0; off >>= 1)
        sum += __shfl_down(sum, off, 32);
    if (lane == 0) s_redB[wave] = sum;
    __syncthreads();
    if (tid == 0) {
        float t = 0.0f;
#pragma unroll
        for (int i = 0; i < ATTN_WAVES; ++i) t += s_redB[i];
        s_stat[1] = 1.0f / t;
    }
    __syncthreads();
    const float inv = s_stat[1];

    // ---- Phase 3: V copy (non-temporal) + weighted accumulation ----
    v4f acc = {0.f, 0.f, 0.f, 0.f};
    for (int k = wave; k <= LCACHE; k += ATTN_WAVES) {
        v4f vr;
        if (k < LCACHE) {
            vr = __builtin_nontemporal_load(
                     reinterpret_cast<const v4f*>(cache_v + srcBase + (size_t)k * DKH) + lane);
            __builtin_nontemporal_store(
                vr, reinterpret_cast<v4f*>(Vout + dstBase + (size_t)k * DKH) + lane);
        } else {
            vr = reinterpret_cast<const v4f*>(Vout + dstBase + (size_t)k * DKH)[lane];
        }
        const float p = s_scores[k];
        acc.x += p * vr.x;
        acc.y += p * vr.y;
        acc.z += p * vr.z;
        acc.w += p * vr.w;
    }
    reinterpret_cast<v4f*>(s_acc[wave])[lane] = acc;
    __syncthreads();
    if (tid < DKH) {
        float o = 0.0f;
#pragma unroll
        for (int w = 0; w < ATTN_WAVES; ++w) o += s_acc[w][tid];
        attn_ws[(size_t)b * D_MODEL + h * DKH + tid] = o * inv;
    }
}

// ---------------------------------------------------------------------------
// Launch:  fused QKV proj -> fused attention (+KV copy) -> output proj.
// Stream ordering provides all dependencies.  No atomics => deterministic.
// ---------------------------------------------------------------------------
extern "C" void kernel_launch(void* const* d_in, const int* in_sizes, int n_in,
                              void* d_out, int out_size, void* d_ws, size_t ws_size,
                              hipStream_t stream) {
    (void)in_sizes; (void)n_in; (void)out_size; (void)ws_size;

    const float* x       = (const float*)d_in[0];   // 16 x 1 x 2048
    const float* cache_k = (const float*)d_in[1];   // 16x16x4096x128
    const float* cache_v = (const float*)d_in[2];
    const float* Wq      = (const float*)d_in[3];
    const float* bq      = (const float*)d_in[4];
    const float* Wk      = (const float*)d_in[5];
    const float* bk      = (const float*)d_in[6];
    const float* Wv      = (const float*)d_in[7];
    const float* bv      = (const float*)d_in[8];
    const float* Wo      = (const float*)d_in[9];
    const float* bo      = (const float*)d_in[10];
    // d_in[11] = num_heads (hardcoded 16)

    float* out  = (float*)d_out;                              // 16 x 2048
    float* Kout = out + (size_t)BATCH * D_MODEL;              // 16x16x4097x128
    float* Vout = Kout + (size_t)BATCH * NHEADS * (LCACHE + 1) * DKH;

    float* Qws     = (float*)d_ws;                            // 16 x 2048
    float* attn_ws = Qws + (size_t)BATCH * D_MODEL;           // 16 x 2048

    // Fused Q/K/V projections (K,V scatter into row L of the out caches)
    qkv_proj_wmma<<<96, 128, 0, stream>>>(x, Wq, bq, Wk, bk, Wv, bv,
                                          Qws, Kout, Vout);

    // Fused cache-copy + attention: one block per (b,h)
    attn_fused<<<BATCH * NHEADS, 32 * ATTN_WAVES, 0, stream>>>(
        cache_k, cache_v, Qws, Kout, Vout, attn_ws);

    // Output projection
    oproj_wmma<<<32, 128, 0, stream>>>(attn_ws, Wo, bo, out);
}